// ForceField_50319836839981
// MI455X (gfx1250) — compile-verified
//
#include <hip/hip_runtime.h>
#include <math.h>

// Problem constants from the reference
constexpr int   Bn   = 4;
constexpr int   Mn   = 4096;         // power of two: n / Mn == n >> 12
constexpr int   Nn   = Bn * Mn;      // 16384
constexpr int   PADI = -999;
#define PADF (-999.0f)
constexpr float CUTOFF2 = 49.0f;     // CUTOFF^2, sqrt is monotone so d2-space compare

typedef float v2f __attribute__((ext_vector_type(2)));
typedef float v8f __attribute__((ext_vector_type(8)));

__device__ __forceinline__ float3 sub3(float3 a, float3 b) {
    return make_float3(a.x - b.x, a.y - b.y, a.z - b.z);
}
__device__ __forceinline__ float3 cross3(float3 a, float3 b) {
    return make_float3(a.y * b.z - a.z * b.y,
                       a.z * b.x - a.x * b.z,
                       a.x * b.y - a.y * b.x);
}
__device__ __forceinline__ float dot3(float3 a, float3 b) {
    return a.x * b.x + a.y * b.y + a.z * b.z;
}

// ---------------------------------------------------------------------------
// Kernel A: per-atom work. Writes out column 0 (dihedral angle) and columns
// 2..7 (partner coords). Pure gather + VALU; bandwidth-trivial.
// ---------------------------------------------------------------------------
__global__ __launch_bounds__(256) void atoms_kernel(
    const float* __restrict__ coords,    // (B, M, 3)
    const int*   __restrict__ cia,       // (N,)
    const int*   __restrict__ partners,  // (N, 2)
    const int*   __restrict__ aidx,      // (N, 4)
    float*       __restrict__ out)       // (N, 8)
{
    int n = blockIdx.x * 256 + threadIdx.x;
    if (n >= Nn) return;
    int b = n >> 12;  // atom_batch[n] == n / M

    // partners_final: where(pvalid, coordinates[b, safe_p], PAD)
    #pragma unroll
    for (int k = 0; k < 2; ++k) {
        int  p  = partners[n * 2 + k];
        bool pv = (p != PADI);
        int  sp = pv ? p : 0;
        const float* pc = coords + ((size_t)b * Mn + sp) * 3;
        out[(size_t)n * 8 + 2 + k * 3 + 0] = pv ? pc[0] : PADF;
        out[(size_t)n * 8 + 2 + k * 3 + 1] = pv ? pc[1] : PADF;
        out[(size_t)n * 8 + 2 + k * 3 + 2] = pv ? pc[2] : PADF;
    }

    // dihedral over gathered coords: coords_g[g] = coordinates[g>>12, cia[g]]
    int a[4];
    #pragma unroll
    for (int t = 0; t < 4; ++t) a[t] = aidx[n * 4 + t];
    bool am = (a[0] != PADI) && (a[1] != PADI) && (a[2] != PADI) && (a[3] != PADI);

    float3 p[4];
    #pragma unroll
    for (int t = 0; t < 4; ++t) {
        int g  = am ? a[t] : 0;
        int gb = g >> 12;
        int gm = cia[g];
        const float* c = coords + ((size_t)gb * Mn + gm) * 3;
        p[t] = make_float3(c[0], c[1], c[2]);
    }
    float3 b1 = sub3(p[1], p[0]);
    float3 b2 = sub3(p[2], p[1]);
    float3 b3 = sub3(p[3], p[2]);
    float3 n1 = cross3(b1, b2);
    float3 n2 = cross3(b2, b3);
    float  inv = 1.0f / (sqrtf(dot3(b2, b2)) + 1e-12f);
    float3 b2n = make_float3(b2.x * inv, b2.y * inv, b2.z * inv);
    float3 m1  = cross3(n1, b2n);
    float  xx  = dot3(n1, n2);
    float  yy  = dot3(m1, n2);
    out[(size_t)n * 8 + 0] = am ? atan2f(yy, xx) : PADF;
}

// ---------------------------------------------------------------------------
// Kernel B: neighbor-distance row sums via f32 WMMA Gram matrix.
// One block = 256 threads = 8 waves; each wave owns one 16-row i-tile and
// sweeps all 256 j-tiles of its batch with V_WMMA_F32_16X16X4_F32.
// LDS holds the whole batch as float4{x,y,z,|c|^2} (64 KB).
// j-loop is unrolled x2 so two WMMAs are in flight and their D->epilogue
// latency overlaps; sqrt uses the raw v_sqrt_f32 (no fixup expansion).
// ---------------------------------------------------------------------------
__global__ __launch_bounds__(256) void nbr_kernel(
    const float* __restrict__ coords,  // (B, M, 3)
    const int*   __restrict__ cia,     // (N,)
    float*       __restrict__ out)     // (N, 8), writes column 1
{
    __shared__ float4 sh[Mn];  // 64 KB

    int b   = blockIdx.x >> 5;   // 32 blocks per batch
    int blk = blockIdx.x & 31;

    // Cooperative fill: gathered coords + squared norm (uniform trip count).
    for (int m = threadIdx.x; m < Mn; m += 256) {
        int idx = cia[b * Mn + m];
        const float* c = coords + ((size_t)b * Mn + idx) * 3;
        float x = c[0], y = c[1], z = c[2];
        sh[m] = make_float4(x, y, z, x * x + y * y + z * z);
    }
    __syncthreads();

    int wave = threadIdx.x >> 5;
    int lane = threadIdx.x & 31;
    int half = lane >> 4;        // 0: K={0,1}, 1: K={2,3}
    int l    = lane & 15;
    int i0   = (blk * 8 + wave) * 16;

    // A operand (16x4 f32): lane l holds row i0+l; half 0 -> (x,y), half 1 -> (z,0)
    float4 arow = sh[i0 + l];
    v2f A;
    A.x = half ? arow.z : arow.x;
    A.y = half ? 0.0f   : arow.y;

    // Loop-invariant sq_i per D-VGPR: row = i0 + 8*half + v
    float h[8];
    #pragma unroll
    for (int v = 0; v < 8; ++v) h[v] = sh[i0 + half * 8 + v].w;

    float acc0[8], acc1[8];
    #pragma unroll
    for (int v = 0; v < 8; ++v) { acc0[v] = 0.0f; acc1[v] = 0.0f; }

    v8f zc = {};  // zero accumulator input
    for (int j0 = 0; j0 < Mn; j0 += 32) {
        float4 brow0 = sh[j0 + l];        // column j0+l
        float4 brow1 = sh[j0 + 16 + l];   // column j0+16+l
        v2f B0, B1;
        B0.x = half ? brow0.z : brow0.x;
        B0.y = half ? 0.0f    : brow0.y;
        B1.x = half ? brow1.z : brow1.x;
        B1.y = half ? 0.0f    : brow1.y;
        float sqj0 = brow0.w;
        float sqj1 = brow1.w;

        v8f G0 = __builtin_amdgcn_wmma_f32_16x16x4_f32(
            false, A, false, B0, (short)0, zc, false, false);
        v8f G1 = __builtin_amdgcn_wmma_f32_16x16x4_f32(
            false, A, false, B1, (short)0, zc, false, false);

        #pragma unroll
        for (int v = 0; v < 8; ++v) {
            float d2 = fmaf(G0[v], -2.0f, h[v] + sqj0);
            float t  = fmaxf(d2, 0.0f) + 1e-12f;
            float s  = __builtin_amdgcn_sqrtf(t);    // raw v_sqrt_f32
            acc0[v] += (t <= CUTOFF2) ? s : 0.0f;    // invalid cols self-mask (d2 ~ 1e6)
        }
        #pragma unroll
        for (int v = 0; v < 8; ++v) {
            float d2 = fmaf(G1[v], -2.0f, h[v] + sqj1);
            float t  = fmaxf(d2, 0.0f) + 1e-12f;
            float s  = __builtin_amdgcn_sqrtf(t);
            acc1[v] += (t <= CUTOFF2) ? s : 0.0f;
        }
    }

    #pragma unroll
    for (int v = 0; v < 8; ++v) acc0[v] += acc1[v];

    // Reduce across the 16 lanes of each half (column reduction of the tile).
    #pragma unroll
    for (int off = 1; off < 16; off <<= 1) {
        #pragma unroll
        for (int v = 0; v < 8; ++v)
            acc0[v] += __shfl_xor(acc0[v], off, 32);
    }

    if (l == 0) {  // lanes 0 and 16 each publish 8 rows
        #pragma unroll
        for (int v = 0; v < 8; ++v) {
            int  row   = i0 + half * 8 + v;
            bool valid = (sh[row].x != PADF);
            out[((size_t)(b * Mn + row)) * 8 + 1] = valid ? acc0[v] : 0.0f;
        }
    }
}

// ---------------------------------------------------------------------------
extern "C" void kernel_launch(void* const* d_in, const int* in_sizes, int n_in,
                              void* d_out, int out_size, void* d_ws, size_t ws_size,
                              hipStream_t stream) {
    (void)in_sizes; (void)n_in; (void)out_size; (void)d_ws; (void)ws_size;
    const float* coords   = (const float*)d_in[0];  // (B, M, 3)
    // d_in[1] = atom_batch: recomputed on device as n >> 12
    const int*   cia      = (const int*)d_in[2];    // (N,)
    const int*   partners = (const int*)d_in[3];    // (N, 2)
    const int*   aidx     = (const int*)d_in[4];    // (N, 4)
    float*       out      = (float*)d_out;          // (N, 8)

    atoms_kernel<<<Nn / 256, 256, 0, stream>>>(coords, cia, partners, aidx, out);
    nbr_kernel<<<Bn * 32, 256, 0, stream>>>(coords, cia, out);
}